// GCN_19378892440057
// MI455X (gfx1250) — compile-verified
//
#include <hip/hip_runtime.h>

// GCN autoencoder for MI455X (gfx1250).
// - Feature matrix (100000 x 128 f32 = 51.2 MB) fits in the 192 MB L2, so the
//   edge gather/scatter phase is L2-bound: global_load_b128 gathers +
//   global_atomic_add_f32 scatters (RMW resolves in L2's atomic units).
// - Dense 128x128 GEMMs use V_WMMA_F32_16X16X4_F32 (exact f32 path; the op is
//   memory-bound, so lower precision buys nothing).
// - bias+ReLU of each GCN layer is fused into the NEXT consumer's A-load
//   (template <PRE>), saving a full 102 MB read+write pass per layer.

#define N_NODES 100000     // divisible by 16 -> no WMMA M-edge masking needed
#define N_EDGES 1600000
#define LATENT  128
#define NC      (N_NODES * LATENT)   // 12.8M

typedef __attribute__((ext_vector_type(2))) float v2f;
typedef __attribute__((ext_vector_type(8))) float v8f;

// ---------------- degree / normalization ----------------
__global__ void k_deg_init(float* __restrict__ deg) {
  int i = blockIdx.x * blockDim.x + threadIdx.x;
  if (i < N_NODES) deg[i] = 1.0f;                       // self-loop contribution
}

__global__ void k_deg_edges(const long long* __restrict__ dst, float* __restrict__ deg) {
  int i = blockIdx.x * blockDim.x + threadIdx.x;
  if (i < N_EDGES) atomicAdd(&deg[(int)dst[i]], 1.0f);
}

__global__ void k_deg_rsqrt(float* __restrict__ deg) {
  int i = blockIdx.x * blockDim.x + threadIdx.x;
  if (i < N_NODES) deg[i] = rsqrtf(deg[i]);             // deg >= 1 always (self-loops)
}

// ---------------- encoder input: T = x @ W_s  (K = 3, VALU is right here) ----------------
__global__ void k_enc_in(const float* __restrict__ x, const float* __restrict__ W,
                         float* __restrict__ T) {
  int i = blockIdx.x * blockDim.x + threadIdx.x;
  if (i >= NC) return;
  int n = i >> 7, c = i & 127;
  const float* xr = x + n * 3;
  T[i] = xr[0] * W[c] + xr[1] * W[LATENT + c] + xr[2] * W[2 * LATENT + c];
}

// ---------------- 128x128 GEMM via f32 WMMA ----------------
// grid.x = N_NODES/16 node tiles, 256 threads = 8 waves, wave w owns output cols [16w,16w+16).
// A-matrix 16x4 f32 layout (ISA 7.12.2): lane m = lane&15; VGPR0 holds K = kb,
// VGPR1 holds K = kb+1 where kb = (lane<16 ? 0 : 2). B 4x16 mirrors it
// (low-K rows lanes 0-15, high-K rows lanes 16-31). C/D: VGPR r -> M = r + (lane<16?0:8).
// PRE:  A element a_k -> relu(a_k + preB[k])   (previous layer's bias+ReLU, fused)
// POST: D element v   -> relu(v + postB[n])    (this layer's bias+ReLU, fused)
template <bool PRE, bool POST>
__global__ void k_gemm_wmma(const float* __restrict__ H, const float* __restrict__ W,
                            const float* __restrict__ preB, const float* __restrict__ postB,
                            float* __restrict__ T) {
  const int wave = threadIdx.x >> 5;
  const int lane = threadIdx.x & 31;
  const int m0   = blockIdx.x * 16;
  const int n0   = wave * 16;
  const int mloc = lane & 15;
  const int kb   = (lane >> 4) * 2;            // 0 or 2
  const int n    = n0 + mloc;                  // this lane's output column

  const float* hrow = H + (size_t)(m0 + mloc) * LATENT;

  v8f acc = {};
  #pragma unroll 4
  for (int k0 = 0; k0 < LATENT; k0 += 4) {
    float a0 = hrow[k0 + kb];
    float a1 = hrow[k0 + kb + 1];
    if (PRE) {                                  // fused bias+ReLU of previous layer
      a0 = fmaxf(a0 + preB[k0 + kb], 0.0f);
      a1 = fmaxf(a1 + preB[k0 + kb + 1], 0.0f);
    }
    v2f a, b;
    a.x = a0; a.y = a1;
    b.x = W[(size_t)(k0 + kb)     * LATENT + n];
    b.y = W[(size_t)(k0 + kb + 1) * LATENT + n];
    acc = __builtin_amdgcn_wmma_f32_16x16x4_f32(
        /*neg_a=*/false, a, /*neg_b=*/false, b,
        /*c_mod=*/(short)0, acc, /*reuse_a=*/false, /*reuse_b=*/false);
  }

  const float bn    = POST ? postB[n] : 0.0f;   // hoisted, one load
  const int   mbase = m0 + ((lane >> 4) << 3);  // +8 for lanes 16..31
  #pragma unroll
  for (int r = 0; r < 8; ++r) {
    float v = acc[r];
    if (POST) v = fmaxf(v + bn, 0.0f);
    T[(size_t)(mbase + r) * LATENT + n] = v;
  }
}

// ---------------- propagation: agg = D^-1/2 (A+I) D^-1/2 T  (raw, no bias) ----------------
// self-loop term initializes agg (also clears the 0xAA-poisoned workspace)
__global__ void k_self_init(const float* __restrict__ dinv, const float* __restrict__ T,
                            float* __restrict__ agg) {
  int i = blockIdx.x * blockDim.x + threadIdx.x;
  if (i >= NC) return;
  float di = dinv[i >> 7];
  agg[i] = di * di * T[i];
}

// one wave per edge; lane handles 4 consecutive channels (float4 gather, 4 f32 atomics)
__global__ void k_edge_scatter(const long long* __restrict__ src, const long long* __restrict__ dst,
                               const float* __restrict__ dinv, const float* __restrict__ T,
                               float* __restrict__ agg) {
  int gid  = blockIdx.x * blockDim.x + threadIdx.x;
  int e    = gid >> 5;
  int lane = threadIdx.x & 31;
  if (e >= N_EDGES) return;
  int s = (int)src[e];
  int d = (int)dst[e];
  float norm = dinv[s] * dinv[d];
  float4 v = ((const float4*)(T + (size_t)s * LATENT))[lane];
  float* o = agg + (size_t)d * LATENT + lane * 4;
  atomicAdd(o + 0, norm * v.x);
  atomicAdd(o + 1, norm * v.y);
  atomicAdd(o + 2, norm * v.z);
  atomicAdd(o + 3, norm * v.w);
}

// ---------------- decoder output: y = h @ W_ed + b_ed + x  (128 -> 3) ----------------
__global__ void k_dec_out(const float* __restrict__ h, const float* __restrict__ Wed,
                          const float* __restrict__ bed, const float* __restrict__ x,
                          float* __restrict__ y) {
  int i = blockIdx.x * blockDim.x + threadIdx.x;
  if (i >= N_NODES * 3) return;
  int node = i / 3, c = i - node * 3;
  const float* hr = h + (size_t)node * LATENT;
  float acc = bed[c];
  #pragma unroll 8
  for (int k = 0; k < LATENT; ++k) acc += hr[k] * Wed[k * 3 + c];
  y[i] = acc + x[i];
}

// ---------------- host orchestration ----------------
extern "C" void kernel_launch(void* const* d_in, const int* in_sizes, int n_in,
                              void* d_out, int out_size, void* d_ws, size_t ws_size,
                              hipStream_t stream) {
  (void)in_sizes; (void)n_in; (void)out_size; (void)ws_size;

  const float*     x    = (const float*)d_in[0];
  const long long* ei   = (const long long*)d_in[1];   // jnp.int64 per reference
  const float*     W_s  = (const float*)d_in[2];   const float* b_s  = (const float*)d_in[3];
  const float*     W_m  = (const float*)d_in[4];   const float* b_m  = (const float*)d_in[5];
  const float*     W_e  = (const float*)d_in[6];   const float* b_e  = (const float*)d_in[7];
  const float*     W_sd = (const float*)d_in[8];   const float* b_sd = (const float*)d_in[9];
  const float*     W_md = (const float*)d_in[10];  const float* b_md = (const float*)d_in[11];
  const float*     W_ed = (const float*)d_in[12];  const float* b_ed = (const float*)d_in[13];
  float*           y    = (float*)d_out;

  const long long* srcI = ei;             // edge_index[0]
  const long long* dstI = ei + N_EDGES;   // edge_index[1]

  // workspace: dinv (0.5 MB) + three 51.2 MB f32 feature buffers
  char* ws = (char*)d_ws;
  const size_t FEAT = (size_t)NC * sizeof(float);
  float* dinv = (float*)ws;
  float* B0   = (float*)(ws + (512u << 10));
  float* B1   = (float*)(ws + (512u << 10) + FEAT);
  float* B2   = (float*)(ws + (512u << 10) + 2 * FEAT);

  const int TB = 256;
  const int gN    = (N_NODES + TB - 1) / TB;     // node-wise
  const int gE    = (N_EDGES + TB - 1) / TB;     // edge-wise
  const int gNC   = NC / TB;                     // 50000, exact
  const int gEW   = (N_EDGES * 32) / TB;         // wave-per-edge, exact
  const int gGEMM = N_NODES / 16;                // 6250, exact
  const int gOUT  = (N_NODES * 3 + TB - 1) / TB;

  // ---- normalization coefficients ----
  k_deg_init <<<gN, TB, 0, stream>>>(dinv);
  k_deg_edges<<<gE, TB, 0, stream>>>(dstI, dinv);
  k_deg_rsqrt<<<gN, TB, 0, stream>>>(dinv);

  // ---- propagation: T -> raw agg (bias+ReLU deferred to consumer) ----
  auto propagate = [&](const float* T, float* agg) {
    k_self_init   <<<gNC, TB, 0, stream>>>(dinv, T, agg);
    k_edge_scatter<<<gEW, TB, 0, stream>>>(srcI, dstI, dinv, T, agg);
  };

  // encoder input layer (K=3)
  k_enc_in<<<gNC, TB, 0, stream>>>(x, W_s, B0);
  propagate(B0, B1);                                                  // agg1 = B1

  // middle GCN layers (shared W_m) + end layer; pre-fused relu(agg + b_prev)
  k_gemm_wmma<true , false><<<gGEMM, TB, 0, stream>>>(B1, W_m, b_s, nullptr, B0);
  propagate(B0, B2);                                                  // agg2 = B2
  k_gemm_wmma<true , false><<<gGEMM, TB, 0, stream>>>(B2, W_m, b_m, nullptr, B0);
  propagate(B0, B1);                                                  // agg3 = B1
  k_gemm_wmma<true , false><<<gGEMM, TB, 0, stream>>>(B1, W_e, b_m, nullptr, B2);
  propagate(B2, B0);                                                  // agg4 = B0

  // decoder dense layers: first fuses pre (b_e) and post (b_sd); rest post-only
  k_gemm_wmma<true , true ><<<gGEMM, TB, 0, stream>>>(B0, W_sd, b_e, b_sd, B1);  // h5 = B1
  k_gemm_wmma<false, true ><<<gGEMM, TB, 0, stream>>>(B1, W_md, nullptr, b_md, B2);  // h6 = B2
  k_gemm_wmma<false, true ><<<gGEMM, TB, 0, stream>>>(B2, W_md, nullptr, b_md, B0);  // h7 = B0

  // output projection + residual
  k_dec_out<<<gOUT, TB, 0, stream>>>(B0, W_ed, b_ed, x, y);
}